// Sggnn_siamese_32693291057655
// MI455X (gfx1250) — compile-verified
//
#include <hip/hip_runtime.h>
#include <hip/hip_bf16.h>

// Problem constants (from reference): B=32, N=4, C=3, H=256, W=128, D=512
#define M_DIM   128          // B*N rows
#define F_DIM   98304        // C*H*W  (K of the GEMM)
#define D_DIM   512          // output features
#define KSPLIT  16
#define KCHUNK  (F_DIM / KSPLIT)   // 6144
#define E_ELEMS (M_DIM * D_DIM)    // 65536 floats per e / per partial

// output layout (flattened tuple): d | w | label
#define D_ELEMS   (32 * 32 * 3 * 512)  // 1,572,864
#define W_ELEMS   (32 * 32 * 3 * 3)    // 9,216
#define L_ELEMS   (32 * 32 * 3)        // 3,072

typedef __attribute__((ext_vector_type(2))) float v2f;
typedef __attribute__((ext_vector_type(8))) float v8f;

// ---------------------------------------------------------------------------
// GEMM: e_partial[ksplit] = x[:, krange] * W[krange, :]   (bias cancels in d)
// One wave computes a 16x16 tile of e via V_WMMA_F32_16X16X4_F32.
// block = 256 threads (8 waves): wave -> M tile, blockIdx.x -> N tile,
// blockIdx.y -> K split. W is read exactly once from HBM; x re-reads hit L2.
// ---------------------------------------------------------------------------
__global__ __launch_bounds__(256) void gemm_e_kernel(
    const float* __restrict__ x,   // (128, 98304) row-major
    const float* __restrict__ Wm,  // (98304, 512) row-major
    float* __restrict__ part)      // (KSPLIT, 128, 512)
{
    const int lane = threadIdx.x & 31;
    const int wv   = threadIdx.x >> 5;   // 0..7 -> M tile
    const int half = lane >> 4;          // 0/1
    const int r    = lane & 15;
    const int ct   = blockIdx.x;         // 0..31 -> N tile (16 cols)
    const int k0   = blockIdx.y * KCHUNK;

    const int row = wv * 16 + r;         // A row handled by this lane
    const int col = ct * 16 + r;         // B/C column handled by this lane

    // A (16x4 f32): lanes 0-15 hold K=0,1 ; lanes 16-31 hold K=2,3
    const float* ap  = x  + (size_t)row * F_DIM + k0 + 2 * half;
    // B (4x16 f32): VGPR0 = row K=half, VGPR1 = row K=2+half
    const float* bp0 = Wm + (size_t)(k0 + half)     * D_DIM + col;
    const float* bp1 = Wm + (size_t)(k0 + 2 + half) * D_DIM + col;

    v8f acc = {};
    for (int k = 0; k < KCHUNK; k += 16) {
#pragma unroll
        for (int kk = 0; kk < 16; kk += 4) {
            v2f a = *(const v2f*)ap;     // b64 load: two consecutive K values
            v2f b;
            b[0] = *bp0;                 // row k+half
            b[1] = *bp1;                 // row k+2+half
            acc = __builtin_amdgcn_wmma_f32_16x16x4_f32(
                false, a, false, b, (short)0, acc, false, false);
            ap  += 4;
            bp0 += 4 * D_DIM;
            bp1 += 4 * D_DIM;
        }
    }

    // C/D 16x16 f32: VGPR v, lanes 0-15 -> M=v, lanes 16-31 -> M=v+8
    float* ep = part + (size_t)blockIdx.y * E_ELEMS
                     + (size_t)(wv * 16 + 8 * half) * D_DIM + col;
#pragma unroll
    for (int v = 0; v < 8; ++v)
        ep[(size_t)v * D_DIM] = acc[v];
}

// ---------------------------------------------------------------------------
// Reduce the KSPLIT partials into e (ws offset KSPLIT*E_ELEMS).
// ---------------------------------------------------------------------------
__global__ __launch_bounds__(256) void reduce_e_kernel(
    const float* __restrict__ part, float* __restrict__ e)
{
    const int tid = blockIdx.x * 256 + threadIdx.x;   // < E_ELEMS
    float s = 0.0f;
#pragma unroll
    for (int p = 0; p < KSPLIT; ++p)
        s += part[(size_t)p * E_ELEMS + tid];
    e[tid] = s;
}

// ---------------------------------------------------------------------------
// d[i,j,0,k,f] = (e[4j+0, f] - e[4*((i+j)%32)+1+k, f])^2
// Bias cancels in the subtraction, so e without bias is exact.
// ---------------------------------------------------------------------------
__global__ __launch_bounds__(256) void d_kernel(
    const float* __restrict__ e, float* __restrict__ out)
{
    const int tid = blockIdx.x * 256 + threadIdx.x;   // < D_ELEMS (exact)
    const int f   = tid & 511;
    const int t   = tid >> 9;          // i*96 + j*3 + k
    const int k   = t % 3;
    const int ij  = t / 3;
    const int j   = ij & 31;
    const int i   = ij >> 5;
    const int idx = (i + j) & 31;
    const float ep   = e[(size_t)(j * 4) * D_DIM + f];
    const float eg   = e[(size_t)(idx * 4 + 1 + k) * D_DIM + f];
    const float diff = ep - eg;
    out[tid] = diff * diff;
}

// ---------------------------------------------------------------------------
// w[i,j,a,k] = (y[4j+1+a] == y[4*((i+j)%32)+1+k])
// label[i,j,0,k] = (y[4j] == y[4*((i+j)%32)+1+k])
// ---------------------------------------------------------------------------
__global__ __launch_bounds__(256) void wl_kernel(
    const int* __restrict__ y, float* __restrict__ out)
{
    const int tid = blockIdx.x * 256 + threadIdx.x;   // 48*256 = 12288 exact
    if (tid < W_ELEMS) {
        const int k  = tid % 3;
        const int t  = tid / 3;
        const int a  = t % 3;
        const int ij = t / 3;
        const int j  = ij & 31;
        const int i  = ij >> 5;
        const int idx = (i + j) & 31;
        out[D_ELEMS + tid] =
            (y[j * 4 + 1 + a] == y[idx * 4 + 1 + k]) ? 1.0f : 0.0f;
    } else {
        const int t  = tid - W_ELEMS;   // < L_ELEMS
        const int k  = t % 3;
        const int ij = t / 3;
        const int j  = ij & 31;
        const int i  = ij >> 5;
        const int idx = (i + j) & 31;
        out[D_ELEMS + W_ELEMS + t] =
            (y[j * 4] == y[idx * 4 + 1 + k]) ? 1.0f : 0.0f;
    }
}

extern "C" void kernel_launch(void* const* d_in, const int* in_sizes, int n_in,
                              void* d_out, int out_size, void* d_ws, size_t ws_size,
                              hipStream_t stream) {
    const float* x  = (const float*)d_in[0];   // (32,4,3,256,128)
    const int*   y  = (const int*)  d_in[1];   // (32,4)
    const float* Wm = (const float*)d_in[2];   // (98304,512)
    // d_in[3] = bias: unused — it cancels in d and is absent from w/label.
    float* out = (float*)d_out;

    float* part = (float*)d_ws;                       // KSPLIT * 256 KB
    float* e    = part + (size_t)KSPLIT * E_ELEMS;    // 256 KB

    // 1) split-K WMMA GEMM into partials (each partial written exactly once,
    //    so no zero-init / atomics needed; deterministic across replays)
    dim3 ggrid(32, KSPLIT);
    gemm_e_kernel<<<ggrid, 256, 0, stream>>>(x, Wm, part);

    // 2) fold partials into e
    reduce_e_kernel<<<E_ELEMS / 256, 256, 0, stream>>>(part, e);

    // 3) pairwise squared differences (dominant output, 6.3 MB)
    d_kernel<<<D_ELEMS / 256, 256, 0, stream>>>(e, out);

    // 4) tiny label tensors
    wl_kernel<<<(W_ELEMS + L_ELEMS) / 256, 256, 0, stream>>>(y, out);
}